// AssociationHead_63642825392169
// MI455X (gfx1250) — compile-verified
//
#include <hip/hip_runtime.h>
#include <math.h>

#define NN 768
#define CH 128

typedef __attribute__((ext_vector_type(2))) float v2f;
typedef __attribute__((ext_vector_type(4))) float v4f;
typedef __attribute__((ext_vector_type(8))) float v8f;

// ---------------------------------------------------------------------------
// Phase 1a: per-node prep.
//   f1[i,:] = relu(W_cls_u[idx[i],:]),  f2[i,:] = relu(W_cls_v[idx[i],:])
//   geometry scalars dx,dy,kx,ky per node.
// ---------------------------------------------------------------------------
__global__ __launch_bounds__(128) void node_prep_kernel(
    const float* __restrict__ loc, const int* __restrict__ cls,
    const float* __restrict__ Wu, const float* __restrict__ Wv,
    float* __restrict__ f1, float* __restrict__ f2,
    float* __restrict__ dxs, float* __restrict__ dys,
    float* __restrict__ kxs, float* __restrict__ kys) {
  const int i = blockIdx.x;
  const int c = threadIdx.x;
  const int k = cls[i];
  f1[i * CH + c] = fmaxf(Wu[k * CH + c], 0.f);
  f2[i * CH + c] = fmaxf(Wv[k * CH + c], 0.f);
  if (c == 0) {
    const float lx = loc[i * 5 + 3];
    const float ly = loc[i * 5 + 4];
    kxs[i] = lx * (1.f / 48.f);
    kys[i] = ly * (1.f / 64.f);
    dxs[i] = (lx - 24.f) * (1.f / 48.f);
    dys[i] = (ly - 32.f) * (1.f / 64.f);
  }
}

// ---------------------------------------------------------------------------
// Phase 1b: vis_feat = relu(vis_maps[768,32] @ W_vis[32,128]) via
// V_WMMA_F32_16X16X4_F32 (exact fp32 — matches reference precision).
// One wave per 16x16 output tile; blockIdx.x = M-tile, wave-in-block = N-tile.
// ---------------------------------------------------------------------------
__global__ __launch_bounds__(256) void vis_gemm_wmma_kernel(
    const float* __restrict__ A,    // [768,32] row-major
    const float* __restrict__ B,    // [32,128] row-major
    float* __restrict__ D) {        // [768,128]
  const int mt   = blockIdx.x * 16;            // 48 M-tiles
  const int nt   = (threadIdx.x >> 5) * 16;    // 8 N-tiles (8 waves/block)
  const int lane = threadIdx.x & 31;
  const int half = lane >> 4;                  // 0: K even pair 0, 1: K pair 2
  const int l16  = lane & 15;

  v8f acc = {};
#pragma unroll
  for (int kk = 0; kk < 32; kk += 4) {
    const int ka = kk + 2 * half;
    // A 16x4 f32 layout: lanes0-15 {V0:K=0,V1:K=1}, lanes16-31 {V0:K=2,V1:K=3}
    v2f a, b;
    a.x = A[(mt + l16) * 32 + ka + 0];
    a.y = A[(mt + l16) * 32 + ka + 1];
    // B 4x16 f32 layout: rows striped across lanes; lanes0-15 rows K=0/1,
    // lanes16-31 rows K=2/3.
    b.x = B[(ka + 0) * CH + nt + l16];
    b.y = B[(ka + 1) * CH + nt + l16];
    acc = __builtin_amdgcn_wmma_f32_16x16x4_f32(
        /*neg_a=*/false, a, /*neg_b=*/false, b,
        /*c_mod=*/(short)0, acc, /*reuse_a=*/false, /*reuse_b=*/false);
  }
  // C/D layout: VGPR r -> row M = r + 8*half, col N = l16.
#pragma unroll
  for (int r = 0; r < 8; ++r) {
    const float v = fmaxf(acc[r], 0.f);
    D[(mt + r + 8 * half) * CH + nt + l16] = v;
  }
}

// ---------------------------------------------------------------------------
// Phase 2: stream feat[i,j,0:384] (906 MB, the bandwidth bound) and fuse the
// W_pred dot + sigmoid. 1 block per row i; 96 threads = 3 wave32s, one wave
// per 128-ch section (0:vis_rel, 1:cls_rel, 2:geo_feat); each lane owns 4
// consecutive channels -> one NT b128 store per lane per (i,j).
// ---------------------------------------------------------------------------
__global__ __launch_bounds__(96) void pair_kernel(
    const float* __restrict__ vis_feat, const float* __restrict__ f1,
    const float* __restrict__ f2,
    const float* __restrict__ dxs, const float* __restrict__ dys,
    const float* __restrict__ kxs, const float* __restrict__ kys,
    const float* __restrict__ W_loc,   // [4,128]
    const float* __restrict__ W_pred,  // [384]
    float* __restrict__ out_ass,       // [768,768]
    float* __restrict__ out_feat) {    // [768,768,384]
  const int t   = threadIdx.x;   // 0..95
  const int i   = blockIdx.x;    // 0..767
  const int sec = t >> 5;        // wave id == section
  const int q   = t & 31;        // lane
  const int c4  = q * 4;         // channel base inside section

  v4f ui = {};                   // per-i vector for vis/cls sections
  v4f wl0 = {}, wl1 = {}, wl2 = {}, wl3 = {};
  if (sec == 0) {
    ui = *(const v4f*)(vis_feat + i * CH + c4);
  } else if (sec == 1) {
    ui = *(const v4f*)(f1 + i * CH + c4);
  } else {
    wl0 = *(const v4f*)(W_loc + 0 * CH + c4);
    wl1 = *(const v4f*)(W_loc + 1 * CH + c4);
    wl2 = *(const v4f*)(W_loc + 2 * CH + c4);
    wl3 = *(const v4f*)(W_loc + 3 * CH + c4);
  }
  const v4f wp = *(const v4f*)(W_pred + sec * CH + c4);
  const float dxi = dxs[i], dyi = dys[i], kxi = kxs[i], kyi = kys[i];

  __shared__ float s_dx[96], s_dy[96], s_kx[96], s_ky[96];
  __shared__ float s_part[3][96];

  for (int jt = 0; jt < NN; jt += 96) {
    s_dx[t] = dxs[jt + t];
    s_dy[t] = dys[jt + t];
    s_kx[t] = kxs[jt + t];
    s_ky[t] = kys[jt + t];
    __syncthreads();

    for (int jj = 0; jj < 96; ++jj) {
      const int j = jt + jj;
      v4f r;
      if (sec == 2) {
        float ddx = dxi - s_dx[jj]; ddx *= ddx;
        float ddy = dyi - s_dy[jj]; ddy *= ddy;
        float dkx = kxi - s_kx[jj]; dkx *= dkx;
        float dky = kyi - s_ky[jj]; dky *= dky;
        r.x = fmaxf(ddx * wl0.x + ddy * wl1.x + dkx * wl2.x + dky * wl3.x, 0.f);
        r.y = fmaxf(ddx * wl0.y + ddy * wl1.y + dkx * wl2.y + dky * wl3.y, 0.f);
        r.z = fmaxf(ddx * wl0.z + ddy * wl1.z + dkx * wl2.z + dky * wl3.z, 0.f);
        r.w = fmaxf(ddx * wl0.w + ddy * wl1.w + dkx * wl2.w + dky * wl3.w, 0.f);
      } else {
        const float* src = (sec == 0) ? vis_feat : f2;
        const v4f vj = *(const v4f*)(src + j * CH + c4);
        r = ui * vj;
      }
      // feat store: 96 lanes x 16B = 1536 contiguous bytes for this (i,j).
      float* fp = out_feat + ((size_t)i * NN + j) * 384 + t * 4;
      __builtin_nontemporal_store(r, (v4f*)fp);

      // fused logit partial: dot with W_pred slice, wave32 reduce.
      float p = r.x * wp.x + r.y * wp.y + r.z * wp.z + r.w * wp.w;
#pragma unroll
      for (int off = 16; off > 0; off >>= 1) p += __shfl_down(p, off, 32);
      if (q == 0) s_part[sec][jj] = p;  // one writer wave per row, no race
    }
    __syncthreads();
    {
      const float s = s_part[0][t] + s_part[1][t] + s_part[2][t];
      const float sig = 1.f / (1.f + __expf(-s));
      __builtin_nontemporal_store(sig, out_ass + (size_t)i * NN + jt + t);
    }
    __syncthreads();  // protect s_dx/s_part before next tile overwrite
  }
}

// ---------------------------------------------------------------------------
extern "C" void kernel_launch(void* const* d_in, const int* in_sizes, int n_in,
                              void* d_out, int out_size, void* d_ws,
                              size_t ws_size, hipStream_t stream) {
  const float* vis_maps  = (const float*)d_in[0];  // [768,32]
  const float* locations = (const float*)d_in[1];  // [768,5]
  const int*   cls       = (const int*)d_in[2];    // [768]
  const float* W_cls_u   = (const float*)d_in[3];  // [17,128]
  const float* W_cls_v   = (const float*)d_in[4];  // [17,128]
  const float* W_loc     = (const float*)d_in[5];  // [4,128]
  const float* W_vis     = (const float*)d_in[6];  // [32,128]
  const float* W_pred    = (const float*)d_in[7];  // [384,1]

  float* ws = (float*)d_ws;
  float* vis_feat = ws;                  // 768*128
  float* f1  = vis_feat + NN * CH;       // 768*128
  float* f2  = f1 + NN * CH;             // 768*128
  float* dxs = f2 + NN * CH;             // 768
  float* dys = dxs + NN;
  float* kxs = dys + NN;
  float* kys = kxs + NN;

  float* out_ass  = (float*)d_out;                 // [768,768]
  float* out_feat = out_ass + (size_t)NN * NN;     // [768,768,384]

  node_prep_kernel<<<NN, CH, 0, stream>>>(locations, cls, W_cls_u, W_cls_v, f1,
                                          f2, dxs, dys, kxs, kys);
  vis_gemm_wmma_kernel<<<48, 256, 0, stream>>>(vis_maps, W_vis, vis_feat);
  pair_kernel<<<NN, 96, 0, stream>>>(vis_feat, f1, f2, dxs, dys, kxs, kys,
                                     W_loc, W_pred, out_ass, out_feat);
}